// MaxConv2d_11940009082933
// MI455X (gfx1250) — compile-verified
//
#include <hip/hip_runtime.h>

#define C_IN   16
#define C_OUT  16
#define H_IN   252
#define W_IN   252
#define H_OUT  253
#define W_OUT  253
#define TILE_W 32
#define TILE_H 8
#define XT_ROWS 11              // TILE_H + 3 halo
#define XT_COLS 36              // TILE_W + 3 halo, padded to 36 (float4-aligned)
#define XT_VALID_COLS 35
#define XT_SIZE (C_IN * XT_ROWS * XT_COLS)   // 6336 floats = 25344 B of LDS

typedef __attribute__((address_space(1))) int g_int_t;  // global
typedef __attribute__((address_space(3))) int l_int_t;  // LDS

// CDNA5 async copy: per-lane global -> LDS, tracked by ASYNCcnt.
// Builtin prototype (from compiler diagnostic): first param is addrspace(1)
// int* (global source), second addrspace(3) int* (LDS dest), then imm offset
// and imm cpol. Generic LDS address low 32 bits == LDS offset (ISA 10.2).
__device__ __forceinline__ void async_copy_b32(const float* gsrc, float* ldst) {
#if __has_builtin(__builtin_amdgcn_global_load_async_to_lds_b32)
  __builtin_amdgcn_global_load_async_to_lds_b32(
      (g_int_t*)(unsigned long long)(const void*)gsrc,
      (l_int_t*)(unsigned)(unsigned long long)(void*)ldst,
      0, 0);
#else
  unsigned laddr = (unsigned)(unsigned long long)(void*)ldst;
  asm volatile("global_load_async_to_lds_b32 %0, %1, off"
               :: "v"(laddr), "v"(gsrc) : "memory");
#endif
}

__device__ __forceinline__ void wait_async_zero() {
#if __has_builtin(__builtin_amdgcn_s_wait_asynccnt)
  __builtin_amdgcn_s_wait_asynccnt(0);
#else
  asm volatile("s_wait_asynccnt 0" ::: "memory");
#endif
}

__global__ __launch_bounds__(256)
void maxconv2d_kernel(const float* __restrict__ x,
                      const float* __restrict__ w,
                      float* __restrict__ out) {
  __shared__ float xt[XT_SIZE];

  const int tid  = threadIdx.x;
  const int lane = tid & 31;
  // Force the wave id to be recognized as uniform so the per-wave weight
  // reads below become scalar (SMEM) loads feeding VALU scalar operands.
  const int wv   = __builtin_amdgcn_readfirstlane(tid >> 5);

  const int tx = blockIdx.x;        // 0..7   (x tiles of 32)
  const int ty = blockIdx.y;        // 0..31  (y tiles of 8)
  const int b  = blockIdx.z;        // 0..3
  const int x0 = tx * TILE_W;       // tile origin in padded coords
  const int y0 = ty * TILE_H;

  // ---- Phase 1: zero the LDS tile (this realizes the zero padding and any
  //      rows/cols beyond the tensor edge). 1584 float4 stores / block.
  float4* xt4 = (float4*)xt;
  for (int t = tid; t < XT_SIZE / 4; t += 256)
    xt4[t] = make_float4(0.f, 0.f, 0.f, 0.f);
  __syncthreads();

  // ---- Phase 2: async-gather the valid input region into LDS.
  //      16 ch x 11 rows x 35 cols = 6160 elements; EXEC-masked per lane.
  const float* xb = x + (size_t)b * C_IN * H_IN * W_IN;
  for (int e = tid; e < C_IN * XT_ROWS * XT_VALID_COLS; e += 256) {
    int c   = e / (XT_ROWS * XT_VALID_COLS);
    int rem = e - c * (XT_ROWS * XT_VALID_COLS);
    int r   = rem / XT_VALID_COLS;
    int col = rem - r * XT_VALID_COLS;
    int iy  = y0 + r - 2;           // padded -> real input row
    int ix  = x0 + col - 2;         // padded -> real input col
    if ((unsigned)iy < (unsigned)H_IN && (unsigned)ix < (unsigned)W_IN) {
      async_copy_b32(xb + ((size_t)c * H_IN + iy) * W_IN + ix,
                     &xt[(c * XT_ROWS + r) * XT_COLS + col]);
    }
  }
  wait_async_zero();
  __syncthreads();

  // ---- Phase 3: compute. Each wave handles 2 output channels; each lane
  //      computes an 8-tall output column for both.
  const int co0 = wv * 2;
  const int co1 = co0 + 1;

  float a0[TILE_H], a1[TILE_H];
#pragma unroll
  for (int y = 0; y < TILE_H; ++y) {
    a0[y] = -3.402823466e+38f;      // finfo(float32).min
    a1[y] = -3.402823466e+38f;
  }

  for (int c = 0; c < C_IN; ++c) {
    // Wave-uniform weight slices -> SGPRs, used as scalar VALU operands.
    float wA[16], wB[16];
#pragma unroll
    for (int ij = 0; ij < 16; ++ij) {
      wA[ij] = w[(co0 * C_IN + c) * 16 + ij];
      wB[ij] = w[(co1 * C_IN + c) * 16 + ij];
    }
    const float* rowbase = &xt[c * XT_ROWS * XT_COLS + lane];
#pragma unroll
    for (int r = 0; r < XT_ROWS; ++r) {
      const float* xr = rowbase + r * XT_COLS;
      float xr0 = xr[0], xr1 = xr[1], xr2 = xr[2], xr3 = xr[3];
#pragma unroll
      for (int i = 0; i < 4; ++i) {
        const int y = r - i;        // output row using kernel row i hits row r
        if (y >= 0 && y < TILE_H) {
          a0[y] = fmaxf(a0[y], xr0 * wA[4 * i + 0]);
          a0[y] = fmaxf(a0[y], xr1 * wA[4 * i + 1]);
          a0[y] = fmaxf(a0[y], xr2 * wA[4 * i + 2]);
          a0[y] = fmaxf(a0[y], xr3 * wA[4 * i + 3]);
          a1[y] = fmaxf(a1[y], xr0 * wB[4 * i + 0]);
          a1[y] = fmaxf(a1[y], xr1 * wB[4 * i + 1]);
          a1[y] = fmaxf(a1[y], xr2 * wB[4 * i + 2]);
          a1[y] = fmaxf(a1[y], xr3 * wB[4 * i + 3]);
        }
      }
    }
  }

  // ---- Phase 4: store (edge-predicated; 253 is not a multiple of the tile).
  const int wox = x0 + lane;
  if (wox < W_OUT) {
    float* ob0 = out + (size_t)(b * C_OUT + co0) * H_OUT * W_OUT + wox;
    float* ob1 = out + (size_t)(b * C_OUT + co1) * H_OUT * W_OUT + wox;
#pragma unroll
    for (int y = 0; y < TILE_H; ++y) {
      const int woy = y0 + y;
      if (woy < H_OUT) {
        ob0[(size_t)woy * W_OUT] = a0[y];
        ob1[(size_t)woy * W_OUT] = a1[y];
      }
    }
  }
}

extern "C" void kernel_launch(void* const* d_in, const int* in_sizes, int n_in,
                              void* d_out, int out_size, void* d_ws, size_t ws_size,
                              hipStream_t stream) {
  const float* x = (const float*)d_in[0];   // (4,16,252,252) f32
  const float* w = (const float*)d_in[1];   // (16,16,4,4)    f32
  float* out = (float*)d_out;               // (4,16,253,253) f32

  dim3 grid((W_OUT + TILE_W - 1) / TILE_W,  // 8
            (H_OUT + TILE_H - 1) / TILE_H,  // 32
            4);                             // batch
  maxconv2d_kernel<<<grid, 256, 0, stream>>>(x, w, out);
}